// ATTPool_60885456388114
// MI455X (gfx1250) — compile-verified
//
#include <hip/hip_runtime.h>
#include <hip/hip_bf16.h>
#include <math.h>

typedef float v2f __attribute__((ext_vector_type(2)));
typedef float v8f __attribute__((ext_vector_type(8)));
typedef float f4  __attribute__((ext_vector_type(4)));

#define B_   8
#define C_   64
#define T_   32
#define HW_  4096           // 64*64
#define HW4_ 1024           // HW_/4 (float4 units)

// Constant positional encoding: 0.1 at odd k in [1,61], else 0.
__device__ __forceinline__ float pos_val(int k) {
    return ((k & 1) && (k < 62)) ? 0.1f : 0.0f;
}

// ---------------------------------------------------------------------------
// Kernel 1: att[row, n] = (x_tre[row,:] + pos) @ W[n,:]^T + bias[n]
// rows = B*T = 256, n = 64, K = 64.  One wave per 16x16 output tile
// (16 M-tiles x 4 N-tiles = 64 waves), K swept as 16 x v_wmma_f32_16x16x4_f32.
// A layout (ISA 7.12.2, 32-bit A 16x4): lanes 0-15 -> K = k0,k0+1;
// lanes 16-31 -> K = k0+2,k0+3 (same M = lane%16 in both halves).
// B layout mirrors A with N = lane%16. C/D: VGPR r -> M = r + 8*(lane/16).
// ---------------------------------------------------------------------------
__global__ void att_gemm_wmma(const float* __restrict__ xt,   // (256, 64)
                              const float* __restrict__ Wm,   // (64, 64)
                              const float* __restrict__ bias, // (64)
                              float* __restrict__ att) {      // (256, 64)
    const int wave  = blockIdx.x * (blockDim.x >> 5) + (threadIdx.x >> 5); // 0..63
    const int tileM = wave >> 2;          // 0..15
    const int tileN = wave & 3;           // 0..3
    const int lane  = threadIdx.x & 31;
    const int half  = lane >> 4;          // 0 or 1
    const int lrow  = lane & 15;
    const int rowA  = tileM * 16 + lrow;  // row of (B*T) x K matrix
    const int colN  = tileN * 16 + lrow;  // output channel

    v8f acc;
    const float bv = bias[colN];
    #pragma unroll
    for (int r = 0; r < 8; ++r) acc[r] = bv;

    #pragma unroll
    for (int k0 = 0; k0 < 64; k0 += 4) {
        const int ka = k0 + 2 * half;
        v2f a, bm;
        a.x  = xt[rowA * 64 + ka]     + pos_val(ka);
        a.y  = xt[rowA * 64 + ka + 1] + pos_val(ka + 1);
        bm.x = Wm[colN * 64 + ka];        // B[k][n] = W[n][k]
        bm.y = Wm[colN * 64 + ka + 1];
        acc = __builtin_amdgcn_wmma_f32_16x16x4_f32(
            /*neg_a=*/false, a, /*neg_b=*/false, bm,
            /*c_mod=*/(short)0, acc, /*reuse_a=*/false, /*reuse_b=*/false);
    }

    #pragma unroll
    for (int r = 0; r < 8; ++r) {
        const int m = tileM * 16 + r + 8 * half;   // global row (b*T + t)
        att[m * 64 + colN] = acc[r];
    }
}

// ---------------------------------------------------------------------------
// Kernel 2: softmax over T per (b,c) column; fold the "+ x[:,:,-1,:,:]"
// residual by adding 1.0 to the t = T-1 weight.  Output wgt is (B, C, T),
// contiguous in t so kernel 3 can broadcast it via LDS.
// ---------------------------------------------------------------------------
__global__ void softmax_t(const float* __restrict__ att,  // (B*T, C)
                          float* __restrict__ wgt) {      // (B*C, T)
    const int idx = blockIdx.x * blockDim.x + threadIdx.x;   // 0..511
    if (idx >= B_ * C_) return;
    const int b = idx >> 6;
    const int c = idx & 63;

    float v[T_];
    float m = -INFINITY;
    #pragma unroll
    for (int t = 0; t < T_; ++t) {
        v[t] = att[(b * T_ + t) * C_ + c];
        m = fmaxf(m, v[t]);
    }
    float s = 0.0f;
    #pragma unroll
    for (int t = 0; t < T_; ++t) {
        v[t] = __expf(v[t] - m);
        s += v[t];
    }
    const float inv = 1.0f / s;
    #pragma unroll
    for (int t = 0; t < T_; ++t) {
        float w = v[t] * inv;
        if (t == T_ - 1) w += 1.0f;     // residual last-slice add folded in
        wgt[idx * T_ + t] = w;
    }
}

// ---------------------------------------------------------------------------
// Kernel 3: out[b,c,h,w] = sum_t wgt[b,c,t] * x[b,c,t,h,w].
// Pure HBM stream (256 MB read once): block = fixed (b,c) + 1/4 of the HW
// plane; each thread reduces one float4 over T with non-temporal b128 loads
// (x is single-use, keep it from thrashing L2), weights broadcast from LDS.
// ---------------------------------------------------------------------------
__global__ void pool_stream(const float* __restrict__ x,    // (B,C,T,H,W)
                            const float* __restrict__ wgt,  // (B*C, T)
                            float* __restrict__ out) {      // (B,C,H,W)
    __shared__ float ws_w[T_];
    const int bc = blockIdx.y;                       // 0..511
    if (threadIdx.x < T_) ws_w[threadIdx.x] = wgt[bc * T_ + threadIdx.x];
    __syncthreads();

    const int pix4 = blockIdx.x * blockDim.x + threadIdx.x;  // 0..1023
    const f4* xp = (const f4*)x + (size_t)bc * (T_ * HW4_) + pix4;

    f4 acc = (f4)0.0f;
    #pragma unroll 8
    for (int t = 0; t < T_; ++t) {
        f4 v = __builtin_nontemporal_load(&xp[(size_t)t * HW4_]);
        acc += ws_w[t] * v;
    }
    __builtin_nontemporal_store(acc, (f4*)out + (size_t)bc * HW4_ + pix4);
}

// ---------------------------------------------------------------------------
extern "C" void kernel_launch(void* const* d_in, const int* in_sizes, int n_in,
                              void* d_out, int out_size, void* d_ws, size_t ws_size,
                              hipStream_t stream) {
    const float* x    = (const float*)d_in[0];  // (8,64,32,64,64)
    const float* xt   = (const float*)d_in[1];  // (8,32,64)
    const float* Wm   = (const float*)d_in[2];  // (64,64)
    const float* bias = (const float*)d_in[3];  // (64)
    float* out = (float*)d_out;                 // (8,64,64,64)

    float* att = (float*)d_ws;                  // 256*64 floats = 64 KB
    float* wgt = att + (B_ * T_ * C_);          // 512*32 floats = 64 KB

    // 1) attention GEMM via WMMA: 64 waves = 8 blocks x 8 waves
    att_gemm_wmma<<<dim3(8), dim3(256), 0, stream>>>(xt, Wm, bias, att);

    // 2) softmax over T (+ residual fold): 512 threads
    softmax_t<<<dim3(2), dim3(256), 0, stream>>>(att, wgt);

    // 3) weighted temporal reduction: (4 chunks) x (B*C = 512 planes)
    pool_stream<<<dim3(HW4_ / 256, B_ * C_), dim3(256), 0, stream>>>(x, wgt, out);
}